// QuantReLU_87290915323937
// MI455X (gfx1250) — compile-verified
//
#include <hip/hip_runtime.h>
#include <stdint.h>

#define BINS 2048
#define TPB  256

// ---------- ordered-float <-> uint key (for global atomic min/max) ----------
__device__ __forceinline__ unsigned fkey(float f) {
  unsigned u = __float_as_uint(f);
  return (u & 0x80000000u) ? ~u : (u | 0x80000000u);
}
__device__ __forceinline__ float fdec(unsigned k) {
  return __uint_as_float((k & 0x80000000u) ? (k & 0x7fffffffu) : ~k);
}

// ws layout (uint words): [0]=minkey [1]=maxkey [2..2+BINS)=hist [2050]=scale(f32) [2051]=zp(f32)

// ---------------- kernel 0: reset workspace (runs every launch) ----------------
__global__ void __launch_bounds__(TPB) init_ws(unsigned* __restrict__ ws) {
  int i = blockIdx.x * blockDim.x + threadIdx.x;
  if (i == 0) { ws[0] = 0xFFFFFFFFu; ws[1] = 0u; }
  for (int j = 2 + i; j < 2 + BINS; j += gridDim.x * blockDim.x) ws[j] = 0u;
}

// ---------------- kernel 1: global min/max (wave32 reduce + atomics) ----------------
__global__ void __launch_bounds__(TPB) minmax_k(const float* __restrict__ x, long long n,
                                                unsigned* __restrict__ ws) {
  long long tid = (long long)blockIdx.x * blockDim.x + threadIdx.x;
  long long stride = (long long)gridDim.x * blockDim.x;
  float mn = __int_as_float(0x7f800000);   // +inf
  float mx = -mn;
  long long n4 = n >> 2;
  const float4* x4 = (const float4*)x;
  for (long long i = tid; i < n4; i += stride) {
    float4 v = x4[i];
    mn = fminf(mn, fminf(fminf(v.x, v.y), fminf(v.z, v.w)));
    mx = fmaxf(mx, fmaxf(fmaxf(v.x, v.y), fmaxf(v.z, v.w)));
  }
  for (long long i = (n4 << 2) + tid; i < n; i += stride) {
    float v = x[i];
    mn = fminf(mn, v);
    mx = fmaxf(mx, v);
  }
#pragma unroll
  for (int off = 16; off; off >>= 1) {
    mn = fminf(mn, __shfl_xor(mn, off, 32));
    mx = fmaxf(mx, __shfl_xor(mx, off, 32));
  }
  if ((threadIdx.x & 31) == 0) {
    atomicMin(&ws[0], fkey(mn));
    atomicMax(&ws[1], fkey(mx));
  }
}

// ---------------- kernel 2: 2048-bin histogram ----------------
// Double-buffered async global->LDS staging (gfx1250 GLOBAL_LOAD_ASYNC_TO_LDS_B128,
// GVS mode, synchronized with s_wait_asynccnt). Each wave stages and consumes its
// own 32x16B slice, so the hot loop needs no block barriers. LDS-private histogram
// via ds_add_u32 atomics, merged once with global atomics.
__global__ void __launch_bounds__(TPB) hist_k(const float* __restrict__ x, long long n,
                                              unsigned* __restrict__ ws) {
  __shared__ unsigned lhist[BINS];
  __shared__ float4 stage[2][TPB];
  const int tid = threadIdx.x;

  for (int i = tid; i < BINS; i += blockDim.x) lhist[i] = 0u;
  __syncthreads();

  const float mn = fdec(ws[0]);
  const float mx = fdec(ws[1]);
  const float width = mx - mn;

  const long long chunkFloats = (long long)TPB * 4;   // 1024 floats / block-chunk
  const long long nChunks = n / chunkFloats;
  const long long cstride = gridDim.x;

  const unsigned long long base = (unsigned long long)(uintptr_t)x;
  const unsigned lds0 = (unsigned)(uintptr_t)&stage[0][tid];
  const unsigned lds1 = (unsigned)(uintptr_t)&stage[1][tid];

  int buf = 0;
  long long cur = blockIdx.x;
  bool havePrev = (cur < nChunks);
  if (havePrev) {
    unsigned voff = (unsigned)((cur * chunkFloats + (long long)tid * 4) * 4);
    asm volatile("global_load_async_to_lds_b128 %0, %1, %2"
                 :: "v"(lds0), "v"(voff), "s"(base) : "memory");
  }
  while (havePrev) {
    long long nxt = cur + cstride;
    bool haveNext = (nxt < nChunks);
    if (haveNext) {
      unsigned voff = (unsigned)((nxt * chunkFloats + (long long)tid * 4) * 4);
      unsigned ldsN = buf ? lds0 : lds1;
      asm volatile("global_load_async_to_lds_b128 %0, %1, %2"
                   :: "v"(ldsN), "v"(voff), "s"(base) : "memory");
      asm volatile("s_wait_asynccnt 1" ::: "memory");   // oldest of 2 complete (in-order)
    } else {
      asm volatile("s_wait_asynccnt 0" ::: "memory");
    }
    float4 v = stage[buf][tid];                          // ds_load_b128
    {
      // faithful: idx = clip(int32((v - mn) / width * 2048), 0, 2047)
      float t0 = (v.x - mn) / width * 2048.0f;
      float t1 = (v.y - mn) / width * 2048.0f;
      float t2 = (v.z - mn) / width * 2048.0f;
      float t3 = (v.w - mn) / width * 2048.0f;
      int b0 = min(max((int)t0, 0), BINS - 1);
      int b1 = min(max((int)t1, 0), BINS - 1);
      int b2 = min(max((int)t2, 0), BINS - 1);
      int b3 = min(max((int)t3, 0), BINS - 1);
      atomicAdd(&lhist[b0], 1u);
      atomicAdd(&lhist[b1], 1u);
      atomicAdd(&lhist[b2], 1u);
      atomicAdd(&lhist[b3], 1u);
    }
    buf ^= 1;
    cur = nxt;
    havePrev = haveNext;
  }

  // scalar tail (n % 1024), handled by block 0
  if (blockIdx.x == 0) {
    for (long long i = nChunks * chunkFloats + tid; i < n; i += blockDim.x) {
      float t = (x[i] - mn) / width * 2048.0f;
      int b = min(max((int)t, 0), BINS - 1);
      atomicAdd(&lhist[b], 1u);
    }
  }
  __syncthreads();
  for (int i = tid; i < BINS; i += blockDim.x)
    if (lhist[i]) atomicAdd(&ws[2 + i], lhist[i]);
}

// ---------------- kernel 3: greedy non-linear param search (single wave32) ----------------
__global__ void __launch_bounds__(32) search_k(unsigned* __restrict__ ws) {
  __shared__ float hist[BINS];
  __shared__ float csum[BINS];
  const int lane = threadIdx.x;
  const float mn = fdec(ws[0]);
  const float mx = fdec(ws[1]);

  for (int i = lane; i < BINS; i += 32) hist[i] = (float)ws[2 + i];
  __syncthreads();
  if (lane == 0) {
    float c = 0.f;
    for (int i = 0; i < BINS; ++i) { c += hist[i]; csum[i] = c; }
  }
  __syncthreads();

  const float total = csum[BINS - 1];
  const float bin_width = (mx - mn) / (float)BINS;

  float alpha = 0.f, beta = 1.f;
  int sb = 0, eb = BINS - 1;
  float norm_min = __int_as_float(0x7f800000);
  bool done = false;

  while (alpha < beta && !done) {
    float na = alpha + 1e-5f;
    float nb = beta - 1e-5f;
    // searchsorted(csum, na*total, 'left'): first i with csum[i] >= v
    int lo = 0, hi = BINS;
    float vl = na * total;
    while (lo < hi) { int m = (lo + hi) >> 1; if (csum[m] < vl) lo = m + 1; else hi = m; }
    int l = max(sb, lo);
    // searchsorted(csum, nb*total, 'right') - 1, clipped to [0, eb]
    lo = 0; hi = BINS;
    float vr = nb * total;
    while (lo < hi) { int m = (lo + hi) >> 1; if (csum[m] <= vr) lo = m + 1; else hi = m; }
    int r = min(max(lo - 1, 0), eb);

    bool move_start = (l - sb) > (eb - r);
    int nsb = move_start ? l : sb;
    int neb = move_start ? eb : r;
    if (move_start) alpha = na; else beta = nb;

    if (nsb != sb || neb != eb) {
      // _compute_quantization_error, parallel over 32 lanes
      float nbinsf = (float)(neb - nsb + 1);
      float dbw = bin_width * nbinsf / 256.0f;
      float norm = 0.f;
      if (dbw != 0.f) {
        float half = dbw / 2.0f;
        float h3 = half * half * half;
        float mhalf = -half;
        float mh3 = mhalf * mhalf * mhalf;
        float acc = 0.f;
        for (int i = lane; i < BINS; i += 32) {
          float src_begin = ((float)i - (float)nsb) * bin_width;
          float src_end = src_begin + bin_width;
          float dob = floorf(src_begin / dbw); dob = fminf(fmaxf(dob, 0.f), 255.f);
          float doe = floorf(src_end / dbw);   doe = fminf(fmaxf(doe, 0.f), 255.f);
          float density = hist[i] / bin_width;
          float begin_center = (dob + 0.5f) * dbw;
          float d0 = src_begin - begin_center;
          float t = density * (h3 - d0 * d0 * d0) / 3.0f;
          t += (doe - dob - 1.0f) * (density * (h3 - mh3) / 3.0f);
          float end_center = doe * dbw + half;
          float d1 = src_end - end_center;
          t += density * (d1 * d1 * d1 - mh3) / 3.0f;
          acc += t;
        }
#pragma unroll
        for (int off = 16; off; off >>= 1) acc += __shfl_xor(acc, off, 32);
        norm = acc;
      }
      bool worse = norm > norm_min;
      if (!worse) { sb = nsb; eb = neb; norm_min = norm; }
      done = worse;
    }
  }

  if (lane == 0) {
    float bw = (mx - mn) / (float)BINS;
    float new_min = mn + bw * (float)sb;
    float new_max = mn + bw * (float)(eb + 1);
    float min_neg = fminf(new_min, 0.f);
    float max_pos = fmaxf(new_max, 0.f);
    float scale = fmaxf((max_pos - min_neg) / 127.0f, 1.1920928955078125e-7f);
    float zp = -64.0f - rintf(min_neg / scale);   // round-half-even, matches jnp.round
    zp = fminf(fmaxf(zp, -64.0f), 63.0f);
    float* wsf = (float*)ws;
    wsf[2 + BINS] = scale;
    wsf[2 + BINS + 1] = zp;
  }
}

// ---------------- kernel 4: quantize (streaming, b128 in / b128 out) ----------------
__global__ void __launch_bounds__(TPB) quant_k(const float* __restrict__ x, long long n,
                                               const float* __restrict__ qp,
                                               float* __restrict__ out) {
  const float scale = qp[0];
  const float zp = qp[1];
  long long tid = (long long)blockIdx.x * blockDim.x + threadIdx.x;
  long long stride = (long long)gridDim.x * blockDim.x;
  long long n4 = n >> 2;
  const float4* x4 = (const float4*)x;
  float4* o4 = (float4*)out;
  for (long long i = tid; i < n4; i += stride) {
    float4 v = x4[i];
    float4 r;
    r.x = (float)(signed char)(int)(v.x / scale + zp);   // trunc toward zero, int8 wrap
    r.y = (float)(signed char)(int)(v.y / scale + zp);
    r.z = (float)(signed char)(int)(v.z / scale + zp);
    r.w = (float)(signed char)(int)(v.w / scale + zp);
    o4[i] = r;
  }
  for (long long i = (n4 << 2) + tid; i < n; i += stride)
    out[i] = (float)(signed char)(int)(x[i] / scale + zp);
}

extern "C" void kernel_launch(void* const* d_in, const int* in_sizes, int n_in,
                              void* d_out, int out_size, void* d_ws, size_t ws_size,
                              hipStream_t stream) {
  (void)n_in; (void)out_size; (void)ws_size;
  const float* x = (const float*)d_in[0];
  long long n = (long long)in_sizes[0];
  unsigned* ws = (unsigned*)d_ws;
  float* out = (float*)d_out;

  init_ws<<<8, TPB, 0, stream>>>(ws);
  minmax_k<<<2048, TPB, 0, stream>>>(x, n, ws);
  hist_k<<<512, TPB, 0, stream>>>(x, n, ws);
  search_k<<<1, 32, 0, stream>>>(ws);
  quant_k<<<2048, TPB, 0, stream>>>(x, n, (const float*)(ws + 2 + BINS), out);
}